// ApplyHomography_53291954209362
// MI455X (gfx1250) — compile-verified
//
#include <hip/hip_runtime.h>

#define NIMG 64          // B*D = 2*32
#define CCH  16
#define HH   256
#define WW   256
#define HWSZ (HH * WW)

typedef float v2f __attribute__((ext_vector_type(2)));
typedef float v4f __attribute__((ext_vector_type(4)));
typedef float v8f __attribute__((ext_vector_type(8)));

// ---------------------------------------------------------------------------
// Kernel 1: grid computation via V_WMMA_F32_16X16X4_F32.
// One WMMA computes X,Y,Z = H @ [x,y,1] for 16 pixels x 2 images:
//   A rows 0..2  = H(n0), rows 8..10 = H(n1), rest zero   (16x4)
//   B cols p     = [x_p, y_p, 1, 0]                        (4x16)
// D layout puts row v in lanes 0-15 / row v+8 in lanes 16-31 of VGPR v,
// so lane l directly owns X,Y,Z of pixel (l%16), image (n0 + l/16).
// ---------------------------------------------------------------------------
__global__ __launch_bounds__(256) void ApplyHomography_grid_wmma_kernel(
    const float* __restrict__ hmat,   // (NIMG, 3, 3)
    float* __restrict__ grid)         // (NIMG, H, W, 2)
{
    const int lane   = threadIdx.x & 31;
    const int waveId = threadIdx.x >> 5;
    const int s      = blockIdx.x * 8 + waveId;  // global 16-pixel strip id

    const int pair  = s >> 12;          // / 4096 strips per image
    const int strip = s & 4095;
    const int h     = strip >> 4;
    const int wb    = (strip & 15) << 4;
    const int n0    = pair * 2;

    const int m  = lane & 15;           // row (A) / column (B) index
    const int hi = lane >> 4;           // 0 -> K=0/1 slots, 1 -> K=2/3 slots

    // ---- A operand: H rows, zero-padded K=3 column, zero rows elsewhere ----
    float a0 = 0.0f, a1 = 0.0f;
    {
        int r = -1, n = n0;
        if (m < 3)                 { r = m;     n = n0;     }
        else if (m >= 8 && m < 11) { r = m - 8; n = n0 + 1; }
        if (r >= 0) {
            const float* Hp = hmat + n * 9 + r * 3;
            if (hi == 0) { a0 = Hp[0]; a1 = Hp[1]; }   // K=0, K=1
            else         { a0 = Hp[2]; a1 = 0.0f; }    // K=2, K=3(pad)
        }
    }

    // ---- B operand: column p = [x, y, 1, 0] ----
    const float xp = (float)(wb + m);
    const float yp = (float)h;
    const float b0 = (hi == 0) ? xp : 1.0f;            // K=0 row: x | K=2 row: 1
    const float b1 = (hi == 0) ? yp : 0.0f;            // K=1 row: y | K=3 row: 0

    v2f A  = { a0, a1 };
    v2f Bv = { b0, b1 };
    v8f Cv = {};
    v8f D = __builtin_amdgcn_wmma_f32_16x16x4_f32(
        /*neg_a=*/false, A, /*neg_b=*/false, Bv,
        /*c_mod=*/(short)0, Cv, /*reuse_a=*/false, /*reuse_b=*/false);

    // lane-local: X=D[0], Y=D[1], Z=D[2] for this lane's (image, pixel)
    const float xw = D[0] / D[2];
    const float yw = D[1] / D[2];
    const float gx = (xw - 127.5f) / 127.5f;
    const float gy = (yw - 127.5f) / 127.5f;

    const int n = n0 + hi;
    float* gp = grid + ((size_t)n * HWSZ + (size_t)h * WW + (wb + m)) * 2;
    v2f g = { gx, gy };
    __builtin_nontemporal_store(g, (v2f*)gp);
}

// ---------------------------------------------------------------------------
// Kernel 2: bandwidth-bound bilinear warp (gather + blend), 4 px/thread,
// grid math amortized over all 16 channels, NT b128 stores.
// ---------------------------------------------------------------------------
__global__ __launch_bounds__(256) void ApplyHomography_warp_kernel(
    const float* __restrict__ hmat,   // (NIMG, 3, 3)
    const float* __restrict__ src,    // (NIMG, C, H, W)
    float* __restrict__ warped)       // (NIMG, C, H, W)
{
    const int tid = threadIdx.x;
    const int qid = blockIdx.x * 256 + tid;
    const int wq  = (qid & 63) << 2;        // first pixel x of this quad
    const int h   = (qid >> 6) & 255;       // row
    const int n   = blockIdx.x >> 6;        // image index: uniform per block -> SGPR

    const float* Hn = hmat + n * 9;
    const float h00 = Hn[0], h01 = Hn[1], h02 = Hn[2];
    const float h10 = Hn[3], h11 = Hn[4], h12 = Hn[5];
    const float h20 = Hn[6], h21 = Hn[7], h22 = Hn[8];

    const float yf = (float)h;
    const float Xr = h01 * yf + h02;
    const float Yr = h11 * yf + h12;
    const float Zr = h21 * yf + h22;

    float wgt[4][4];
    int   off[4][4];

    #pragma unroll
    for (int p = 0; p < 4; ++p) {
        const float xf = (float)(wq + p);
        const float X = h00 * xf + Xr;
        const float Y = h10 * xf + Yr;
        const float Z = h20 * xf + Zr;
        const float xw = X / Z;
        const float yw = Y / Z;
        const float gx = (xw - 127.5f) / 127.5f;
        const float gy = (yw - 127.5f) / 127.5f;
        const float ix = ((gx + 1.0f) * (float)WW - 1.0f) * 0.5f;
        const float iy = ((gy + 1.0f) * (float)HH - 1.0f) * 0.5f;
        const float x0f = floorf(ix), y0f = floorf(iy);
        const float wx1 = ix - x0f, wx0 = 1.0f - wx1;
        const float wy1 = iy - y0f, wy0 = 1.0f - wy1;
        const int x0 = (int)x0f, y0 = (int)y0f;
        const int x1 = x0 + 1,   y1 = y0 + 1;
        const bool mx0 = (x0 >= 0) & (x0 < WW);
        const bool mx1 = (x1 >= 0) & (x1 < WW);
        const bool my0 = (y0 >= 0) & (y0 < HH);
        const bool my1 = (y1 >= 0) & (y1 < HH);
        const int xc0 = min(max(x0, 0), WW - 1), xc1 = min(max(x1, 0), WW - 1);
        const int yc0 = min(max(y0, 0), HH - 1), yc1 = min(max(y1, 0), HH - 1);
        wgt[p][0] = (my0 & mx0) ? wy0 * wx0 : 0.0f;
        wgt[p][1] = (my0 & mx1) ? wy0 * wx1 : 0.0f;
        wgt[p][2] = (my1 & mx0) ? wy1 * wx0 : 0.0f;
        wgt[p][3] = (my1 & mx1) ? wy1 * wx1 : 0.0f;
        off[p][0] = yc0 * WW + xc0;
        off[p][1] = yc0 * WW + xc1;
        off[p][2] = yc1 * WW + xc0;
        off[p][3] = yc1 * WW + xc1;
    }

    const float* sp = src    + (size_t)n * CCH * HWSZ;
    float*       op = warped + (size_t)n * CCH * HWSZ + (size_t)h * WW + wq;

    #pragma unroll 4
    for (int c = 0; c < CCH; ++c) {
        const float* __restrict__ spc = sp + c * HWSZ;
        if (c + 1 < CCH) {
            __builtin_prefetch(spc + HWSZ + off[0][0], 0, 0);   // global_prefetch
        }
        v4f r;
        #pragma unroll
        for (int p = 0; p < 4; ++p) {
            float acc  = spc[off[p][0]] * wgt[p][0];
            acc       += spc[off[p][1]] * wgt[p][1];
            acc       += spc[off[p][2]] * wgt[p][2];
            acc       += spc[off[p][3]] * wgt[p][3];
            r[p] = acc;
        }
        __builtin_nontemporal_store(r, (v4f*)(op + c * HWSZ));
    }
}

extern "C" void kernel_launch(void* const* d_in, const int* in_sizes, int n_in,
                              void* d_out, int out_size, void* d_ws, size_t ws_size,
                              hipStream_t stream) {
    const float* hmat = (const float*)d_in[0];   // (B,D,3,3) = (64,3,3)
    const float* src  = (const float*)d_in[1];   // (B,D,C,H,W)
    float* warped = (float*)d_out;
    float* grid   = (float*)d_out + (size_t)NIMG * CCH * HWSZ;

    // Grid output via WMMA einsum: 32 image pairs * 4096 strips / 8 waves per block
    dim3 block(256);
    dim3 gGrid((NIMG / 2) * (HWSZ / 16) / 8);    // 16384 blocks
    ApplyHomography_grid_wmma_kernel<<<gGrid, block, 0, stream>>>(hmat, grid);

    // Warped output: bandwidth-bound gather/blend
    dim3 gWarp(NIMG * HH / 4);                   // 4096 blocks
    ApplyHomography_warp_kernel<<<gWarp, block, 0, stream>>>(hmat, src, warped);
}